// Caps_Layer_15453292331407
// MI455X (gfx1250) — compile-verified
//
#include <hip/hip_runtime.h>
#include <hip/hip_bf16.h>

// ---------------------------------------------------------------------------
// Capsule routing layer, algebraically refactored to avoid materializing the
// 128MB u_hat tensor.  All heavy math runs through V_WMMA_F32_16X16X4_F32
// (exact fp32 — problem is memory-bound on MI455X, so no precision tradeoff).
// ---------------------------------------------------------------------------

typedef __attribute__((ext_vector_type(2))) float v2f;
typedef __attribute__((ext_vector_type(8))) float v8f;

namespace caps {
constexpr int B   = 32;
constexpr int In  = 4096;
constexpr int D   = 256;
constexpr int N   = 16;   // num_capsule
constexpr int K   = 16;   // dim_capsule
constexpr int SPLIT = 8;  // K-split of the In dimension for the s-GEMM
}

using namespace caps;

// ---------------------------------------------------------------------------
__global__ void zero_buf(float* p, int n) {
    int i = blockIdx.x * blockDim.x + threadIdx.x;
    if (i < n) p[i] = 0.0f;
}

// softmax over the capsule axis; bc layout [B, In, N] so each thread owns a
// contiguous 64B row -> fully coalesced across the wave.  In-place (read then
// write own 16 floats).
__global__ void softmax_n(float* __restrict__ bc) {
    int idx = blockIdx.x * blockDim.x + threadIdx.x;   // (b*In + i)
    if (idx >= B * In) return;
    float* p = bc + (size_t)idx * N;
    float v[N];
    float m = -3.402823466e+38f;
#pragma unroll
    for (int n = 0; n < N; ++n) { v[n] = p[n]; m = fmaxf(m, v[n]); }
    float s = 0.0f;
#pragma unroll
    for (int n = 0; n < N; ++n) { v[n] = __expf(v[n] - m); s += v[n]; }
    float inv = 1.0f / s;
#pragma unroll
    for (int n = 0; n < N; ++n) p[n] = v[n] * inv;
}

// ---------------------------------------------------------------------------
// s_part[b, sp, n, d] = sum_{i in chunk sp} c[b,i,n] * x[b,i,d]
// Per-batch GEMM: M=16 (capsules), N=256 (d), K=In/SPLIT per workgroup.
// Block = 128 threads = 4 waves; wave w owns d-columns [64w, 64w+64) as four
// 16x16 accumulator tiles.  grid = (SPLIT, B).
__global__ void __launch_bounds__(128)
gemm_s(const float* __restrict__ x, const float* __restrict__ c,
       float* __restrict__ s_part) {
    const int b    = blockIdx.y;
    const int sp   = blockIdx.x;
    const int lane = threadIdx.x & 31;
    const int wave = threadIdx.x >> 5;
    const int half = lane >> 4;     // 0: lanes 0-15, 1: lanes 16-31
    const int l16  = lane & 15;

    const float* cb = c + (size_t)b * In * N;   // [In][16]
    const float* xb = x + (size_t)b * In * D;   // [In][256]

    const int d0 = wave * 64;
    const int i_beg = sp * (In / SPLIT);
    const int i_end = i_beg + (In / SPLIT);

    v8f acc0 = {}, acc1 = {}, acc2 = {}, acc3 = {};

    for (int i0 = i_beg; i0 < i_end; i0 += 4) {
        const int kr = i0 + half * 2;           // this half-wave's two K rows
        // A fragment (16x4): A[m][k] = c[b, i0+k, m]
        v2f a;
        a.x = cb[(size_t)(kr + 0) * N + l16];
        a.y = cb[(size_t)(kr + 1) * N + l16];
        // B fragments (4x16 each): B[k][n] = x[b, i0+k, d0 + nt*16 + n]
        const float* xr0 = xb + (size_t)(kr + 0) * D;
        const float* xr1 = xb + (size_t)(kr + 1) * D;
        __builtin_prefetch(xr0 + 4 * D, 0, 1);  // global_prefetch_b8 next chunk
        v2f b0, b1, b2, b3;
        b0.x = xr0[d0 +  0 + l16]; b0.y = xr1[d0 +  0 + l16];
        b1.x = xr0[d0 + 16 + l16]; b1.y = xr1[d0 + 16 + l16];
        b2.x = xr0[d0 + 32 + l16]; b2.y = xr1[d0 + 32 + l16];
        b3.x = xr0[d0 + 48 + l16]; b3.y = xr1[d0 + 48 + l16];
        acc0 = __builtin_amdgcn_wmma_f32_16x16x4_f32(false, a, false, b0, (short)0, acc0, false, false);
        acc1 = __builtin_amdgcn_wmma_f32_16x16x4_f32(false, a, false, b1, (short)0, acc1, false, false);
        acc2 = __builtin_amdgcn_wmma_f32_16x16x4_f32(false, a, false, b2, (short)0, acc2, false, false);
        acc3 = __builtin_amdgcn_wmma_f32_16x16x4_f32(false, a, false, b3, (short)0, acc3, false, false);
    }

    // C/D layout: VGPR r -> row (r + 8*half), col l16 (ISA 7.12.2)
    float* outp = s_part + (size_t)(b * SPLIT + sp) * N * D;
#pragma unroll
    for (int r = 0; r < 8; ++r) {
        const int row = r + half * 8;
        outp[(size_t)row * D + d0 +  0 + l16] = acc0[r];
        outp[(size_t)row * D + d0 + 16 + l16] = acc1[r];
        outp[(size_t)row * D + d0 + 32 + l16] = acc2[r];
        outp[(size_t)row * D + d0 + 48 + l16] = acc3[r];
    }
}

// s[b,n,d] = sum_sp s_part[b,sp,n,d]
__global__ void reduce_s(const float* __restrict__ s_part, float* __restrict__ s) {
    int idx = blockIdx.x * blockDim.x + threadIdx.x;
    if (idx >= B * N * D) return;
    int b  = idx / (N * D);
    int nd = idx - b * (N * D);
    float acc = 0.0f;
#pragma unroll
    for (int sp = 0; sp < SPLIT; ++sp)
        acc += s_part[(size_t)(b * SPLIT + sp) * N * D + nd];
    s[idx] = acc;
}

// outputs[b,n,k] = squash_k( sum_d s[b,n,d] * W[d, n*16+k] )
// grid = B*N blocks of 256 threads (one thread per d).
__global__ void __launch_bounds__(256)
outputs_squash(const float* __restrict__ s, const float* __restrict__ W,
               float* __restrict__ out) {
    __shared__ float part[D * K];   // 16 KB
    __shared__ float dot[K];
    const int bn = blockIdx.x;
    const int n  = bn & (N - 1);
    const int d  = threadIdx.x;

    const float  sv = s[(size_t)bn * D + d];
    const float* wr = W + (size_t)d * (N * K) + (size_t)n * K;
#pragma unroll
    for (int k = 0; k < K; ++k) part[d * K + k] = sv * wr[k];
    __syncthreads();

    if (d < K) {
        float a = 0.0f;
        for (int dd = 0; dd < D; ++dd) a += part[dd * K + d];
        dot[d] = a;
    }
    __syncthreads();

    if (d < K) {
        float n2 = 0.0f;
#pragma unroll
        for (int k = 0; k < K; ++k) n2 += dot[k] * dot[k];
        out[(size_t)bn * K + d] = dot[d] * rsqrtf(n2 + 1e-7f);
    }
}

// t[b,n,d] = sum_k W[d, n*16+k] * outputs[b,n,k]
__global__ void __launch_bounds__(256)
compute_t(const float* __restrict__ W, const float* __restrict__ out,
          float* __restrict__ t) {
    __shared__ float ov[K];
    const int bn = blockIdx.x;
    const int n  = bn & (N - 1);
    const int d  = threadIdx.x;
    if (d < K) ov[d] = out[(size_t)bn * K + d];
    __syncthreads();
    const float* wr = W + (size_t)d * (N * K) + (size_t)n * K;
    float acc = 0.0f;
#pragma unroll
    for (int k = 0; k < K; ++k) acc += wr[k] * ov[k];
    t[(size_t)bn * D + d] = acc;
}

// ---------------------------------------------------------------------------
// b_new[b,i,n] = sum_d x[b,i,d] * t[b,n,d]
// Per-batch GEMM: M=4096 (i), N=16 (capsules), K=256.
// Block = 128 threads = 4 waves, each wave one 16x16 M-tile. grid=(In/64, B).
__global__ void __launch_bounds__(128)
bupdate(const float* __restrict__ x, const float* __restrict__ t,
        float* __restrict__ bc) {
    const int b    = blockIdx.y;
    const int lane = threadIdx.x & 31;
    const int wave = threadIdx.x >> 5;
    const int half = lane >> 4;
    const int l16  = lane & 15;
    const int i0   = (blockIdx.x * 4 + wave) * 16;

    const float* xb = x + (size_t)b * In * D;
    const float* tb = t + (size_t)b * N * D;

    v8f acc = {};
    for (int d0 = 0; d0 < D; d0 += 4) {
        const int kd = d0 + half * 2;
        // A[m][k] = x[b, i0+m, d0+k]
        const float* xr = xb + (size_t)(i0 + l16) * D + kd;
        __builtin_prefetch(xr + 64, 0, 1);
        v2f a;  a.x = xr[0];  a.y = xr[1];
        // B[k][n] = t[b, n, d0+k]
        const float* tr = tb + (size_t)l16 * D + kd;
        v2f bb; bb.x = tr[0]; bb.y = tr[1];
        acc = __builtin_amdgcn_wmma_f32_16x16x4_f32(false, a, false, bb, (short)0, acc, false, false);
    }

    float* dst = bc + ((size_t)b * In + i0) * N;
#pragma unroll
    for (int r = 0; r < 8; ++r)
        dst[(size_t)(r + half * 8) * N + l16] = acc[r];
}

// ---------------------------------------------------------------------------
extern "C" void kernel_launch(void* const* d_in, const int* in_sizes, int n_in,
                              void* d_out, int out_size, void* d_ws, size_t ws_size,
                              hipStream_t stream) {
    const float* x = (const float*)d_in[0];   // [B, In, D]
    const float* W = (const float*)d_in[1];   // [1, D, N*K]
    float* out = (float*)d_out;               // [B, N, K]

    // Workspace carve-up (floats):
    float* ws     = (float*)d_ws;
    float* bc     = ws;                                   // B*In*N   (b / c in-place)
    float* s_part = bc + (size_t)B * In * N;              // B*SPLIT*N*D
    float* s      = s_part + (size_t)B * SPLIT * N * D;   // B*N*D
    float* t      = s + (size_t)B * N * D;                // B*N*D

    const int bc_n = B * In * N;
    zero_buf<<<(bc_n + 255) / 256, 256, 0, stream>>>(bc, bc_n);

    for (int it = 0; it < 3; ++it) {
        softmax_n<<<(B * In + 255) / 256, 256, 0, stream>>>(bc);
        gemm_s<<<dim3(SPLIT, B), 128, 0, stream>>>(x, bc, s_part);
        reduce_s<<<(B * N * D + 255) / 256, 256, 0, stream>>>(s_part, s);
        outputs_squash<<<B * N, 256, 0, stream>>>(s, W, out);
        if (it < 2) {
            compute_t<<<B * N, 256, 0, stream>>>(W, out, t);
            bupdate<<<dim3(In / 64, B), 128, 0, stream>>>(x, t, bc);
        }
    }
}